// CREDALoss_74062416052317
// MI455X (gfx1250) — compile-verified
//
#include <hip/hip_runtime.h>
#include <hip/hip_bf16.h>
#include <math.h>

#define NS 1024
#define NT 1024
#define NN 2048
#define DD 2048
#define NC 4
#define EPSF 1e-8f
#define EPSD 1e-8

typedef __attribute__((ext_vector_type(2))) float v2f;
typedef __attribute__((ext_vector_type(8))) float v8f;

struct Scratch {
    double S1[NC], S2[NC], S3[NC], trss[NC], trtt[NC];
    double cls_sum, ent_sum;
    unsigned lo[NC], hi[NC], cnt_tmp[NC], paircnt[NC], ktarget[NC];
    unsigned cnt_s[NC], cnt_t[NC];
    float invden[NC];
    float x2[NN];
    float wfull[NN];
    int   cls[NN];
};

// ---------------- init ----------------
__global__ void init_kernel(Scratch* sc) {
    int t = threadIdx.x;
    if (t < NC) {
        sc->S1[t] = 0.0; sc->S2[t] = 0.0; sc->S3[t] = 0.0;
        sc->trss[t] = 0.0; sc->trtt[t] = 0.0;
        sc->cnt_tmp[t] = 0u; sc->cnt_s[t] = 0u; sc->cnt_t[t] = 0u;
    }
    if (t == 0) { sc->cls_sum = 0.0; sc->ent_sum = 0.0; }
}

// ------------- per-row logits work: cls loss, entropy, pseudo, w -------------
__global__ void row_kernel(const float* __restrict__ logits_s,
                           const int*   __restrict__ labels_s,
                           const float* __restrict__ logits_t,
                           Scratch* sc) {
    int i = blockIdx.x * blockDim.x + threadIdx.x;
    if (i >= NN) return;
    if (i < NS) {
        const float* l = logits_s + (size_t)i * NC;
        float m = l[0];
        for (int c = 1; c < NC; ++c) m = fmaxf(m, l[c]);
        float z = 0.f;
        for (int c = 0; c < NC; ++c) z += expf(l[c] - m);
        float lse = m + logf(z);
        int lab = labels_s[i];
        float lp = l[lab] - lse;
        atomicAdd(&sc->cls_sum, (double)(-lp));
        sc->cls[i] = lab;
        atomicAdd(&sc->cnt_s[lab], 1u);
        sc->wfull[i] = 0.f;
    } else {
        int t = i - NS;
        const float* l = logits_t + (size_t)t * NC;
        float m = l[0]; int am = 0;
        for (int c = 1; c < NC; ++c) if (l[c] > m) { m = l[c]; am = c; }
        float e[NC]; float z = 0.f;
        for (int c = 0; c < NC; ++c) { e[c] = expf(l[c] - m); z += e[c]; }
        float ent = 0.f;
        for (int c = 0; c < NC; ++c) { float p = e[c] / z; ent -= p * logf(p + EPSF); }
        atomicAdd(&sc->ent_sum, (double)ent);
        sc->cls[i] = am;
        atomicAdd(&sc->cnt_t[am], 1u);
        sc->wfull[i] = 1.f - ent / (logf((float)NC) + EPSF);
    }
}

// ---------------- row squared norms ----------------
__global__ void rownorm_kernel(const float* __restrict__ fs,
                               const float* __restrict__ ft,
                               Scratch* sc) {
    int row = blockIdx.x;
    const float* p = (row < NS) ? (fs + (size_t)row * DD)
                                : (ft + (size_t)(row - NS) * DD);
    float acc = 0.f;
    for (int d = threadIdx.x; d < DD; d += blockDim.x) { float v = p[d]; acc += v * v; }
    __shared__ float red[256];
    red[threadIdx.x] = acc;
    __syncthreads();
    for (int s = 128; s > 0; s >>= 1) {
        if (threadIdx.x < s) red[threadIdx.x] += red[threadIdx.x + s];
        __syncthreads();
    }
    if (threadIdx.x == 0) sc->x2[row] = red[0];
}

// ---- Gram + D2 via f32 WMMA: symmetric, only bi<=bj tile pairs (528 wave-tiles),
// ---- transposed block re-emitted coalesced through a padded per-wave LDS buffer.
__global__ __launch_bounds__(128) void gram_kernel(const float* __restrict__ fs,
                                                   const float* __restrict__ ft,
                                                   float* __restrict__ D2,
                                                   const Scratch* __restrict__ sc) {
    __shared__ float ldsT[4][64 * 65];  // 64x64 tile, padded stride 65 (bank-conflict free)
    int wave = threadIdx.x >> 5;
    int lane = threadIdx.x & 31;
    int tile = blockIdx.x * 4 + wave;   // 0..527 (132 blocks * 4 waves)

    // decode triangular pair bi <= bj  (uniform per wave)
    int bi = 0, rem = tile;
    while (rem >= 32 - bi) { rem -= 32 - bi; ++bi; }
    int bj = bi + rem;

    int rowBase = bi * 64, colBase = bj * 64;
    // 64-row groups never straddle the 1024 source/target boundary
    const float* A = (rowBase < NS) ? (fs + (size_t)rowBase * DD)
                                    : (ft + (size_t)(rowBase - NS) * DD);
    const float* B = (colBase < NS) ? (fs + (size_t)colBase * DD)
                                    : (ft + (size_t)(colBase - NS) * DD);
    int half = lane >> 4, l16 = lane & 15;

    v8f acc[4][4] = {};
    for (int kk = 0; kk < DD; kk += 4) {
        int kb = kk + half * 2;  // lanes 0-15: K=kk,kk+1 ; lanes 16-31: K=kk+2,kk+3
        v2f a[4], b[4];
#pragma unroll
        for (int r = 0; r < 4; ++r)
            a[r] = *(const v2f*)(A + (size_t)(r * 16 + l16) * DD + kb);
#pragma unroll
        for (int c = 0; c < 4; ++c)
            b[c] = *(const v2f*)(B + (size_t)(c * 16 + l16) * DD + kb);
#pragma unroll
        for (int r = 0; r < 4; ++r)
#pragma unroll
            for (int c = 0; c < 4; ++c)
                acc[r][c] = __builtin_amdgcn_wmma_f32_16x16x4_f32(
                    false, a[r], false, b[c], (short)0, acc[r][c], false, false);
    }

    float* T = ldsT[wave];
#pragma unroll
    for (int r = 0; r < 4; ++r)
#pragma unroll
        for (int c = 0; c < 4; ++c)
#pragma unroll
            for (int v = 0; v < 8; ++v) {
                int lr = r * 16 + v + 8 * half;   // local row 0..63
                int lc = c * 16 + l16;            // local col 0..63
                int row = rowBase + lr;
                int col = colBase + lc;
                float g = acc[r][c][v];
                float d2 = fmaxf(sc->x2[row] + sc->x2[col] - 2.f * g, 0.f);
                D2[(size_t)row * NN + col] = d2;          // coalesced direct store
                if (bi != bj) T[lc * 65 + lr] = d2;       // stage transpose in LDS
            }
    __syncthreads();  // all waves reach this (tile indices are wave-uniform)
    if (bi != bj) {
        // transposed 64x64 block at (colBase.., rowBase..), coalesced stores
#pragma unroll 4
        for (int p = 0; p < 64; ++p) {
#pragma unroll
            for (int h = 0; h < 2; ++h) {
                int q = lane + 32 * h;
                D2[(size_t)(colBase + p) * NN + (rowBase + q)] = T[p * 65 + q];
            }
        }
    }
}

// ---------------- median: setup ----------------
__global__ void med_setup_kernel(Scratch* sc) {
    int c = threadIdx.x;
    if (c < NC) {
        unsigned m = sc->cnt_s[c] + sc->cnt_t[c];
        unsigned pc = m * (m - 1u) / 2u;
        sc->paircnt[c] = pc;
        unsigned k = (pc > 0u) ? (pc - 1u) / 2u : 0u;  // lower-median 0-based rank
        sc->ktarget[c] = k + 1u;
        sc->lo[c] = 0u; sc->hi[c] = 0xFFFFFFFFu;
        sc->cnt_tmp[c] = 0u;
    }
}

// ---------------- median: per-class count of (bits <= mid) over upper-tri pairs ----------------
__global__ void med_count_kernel(const float* __restrict__ D2, Scratch* sc) {
    __shared__ unsigned scnt[NC], slo[NC], shi[NC];
    int t = threadIdx.x;
    if (t < NC) { scnt[t] = 0u; slo[t] = sc->lo[t]; shi[t] = sc->hi[t]; }
    __syncthreads();
    size_t idx = (size_t)blockIdx.x * blockDim.x + t;
    int i = (int)(idx >> 11), j = (int)(idx & (NN - 1));
    if (i < j) {
        int ci = sc->cls[i];
        if (sc->cls[j] == ci) {
            unsigned mid = slo[ci] + ((shi[ci] - slo[ci]) >> 1);
            unsigned bits = __float_as_uint(D2[idx]);  // D2 >= 0 -> uint order == float order
            if (bits <= mid) atomicAdd(&scnt[ci], 1u);
        }
    }
    __syncthreads();
    if (t < NC && scnt[t]) atomicAdd(&sc->cnt_tmp[t], scnt[t]);
}

__global__ void med_update_kernel(Scratch* sc) {
    int c = threadIdx.x;
    if (c < NC) {
        unsigned lo = sc->lo[c], hi = sc->hi[c];
        unsigned mid = lo + ((hi - lo) >> 1);
        if (sc->cnt_tmp[c] >= sc->ktarget[c]) sc->hi[c] = mid;
        else sc->lo[c] = mid + 1u;
        sc->cnt_tmp[c] = 0u;
    }
}

__global__ void med_final_kernel(Scratch* sc) {
    int c = threadIdx.x;
    if (c < NC) {
        float med = (sc->paircnt[c] > 0u) ? __uint_as_float(sc->lo[c]) : 1.0f;
        sc->invden[c] = 1.0f / (2.0f * (med + 1e-6f) + EPSF);
    }
}

// ---------------- kernel-sum pass: S1/S2/S3 + traces ----------------
__global__ void sum_kernel(const float* __restrict__ D2, Scratch* sc) {
    __shared__ double s1[NC], s2[NC], s3[NC], ts[NC], tt[NC];
    int t = threadIdx.x;
    if (t < NC) { s1[t] = 0; s2[t] = 0; s3[t] = 0; ts[t] = 0; tt[t] = 0; }
    __syncthreads();
    size_t idx = (size_t)blockIdx.x * blockDim.x + t;
    int i = (int)(idx >> 11), j = (int)(idx & (NN - 1));
    int ci = sc->cls[i];
    if (sc->cls[j] == ci) {
        float k = expf(-D2[idx] * sc->invden[ci]);
        float k2 = k * k;
        bool is_ = (i < NS), js_ = (j < NS);
        if (is_ && js_) {
            atomicAdd(&s1[ci], (double)k2);                       // sum(Kss^2)
        } else if (!is_ && !js_) {
            float ww = sc->wfull[i] * sc->wfull[j];
            atomicAdd(&s2[ci], (double)(ww * ww) * (double)k2);   // sum(Ktt^2)
        } else {
            atomicAdd(&s3[ci], (double)k2);                       // both orderings -> 2*sum(Kst^2)
        }
        if (i == j) {
            if (is_) atomicAdd(&ts[ci], (double)k);               // tr(Kss)
            else {
                float w = sc->wfull[i];
                atomicAdd(&tt[ci], (double)(w * w) * (double)k);  // tr(Ktt)
            }
        }
    }
    __syncthreads();
    if (t < NC) {
        if (s1[t] != 0.0) atomicAdd(&sc->S1[t], s1[t]);
        if (s2[t] != 0.0) atomicAdd(&sc->S2[t], s2[t]);
        if (s3[t] != 0.0) atomicAdd(&sc->S3[t], s3[t]);
        if (ts[t] != 0.0) atomicAdd(&sc->trss[t], ts[t]);
        if (tt[t] != 0.0) atomicAdd(&sc->trtt[t], tt[t]);
    }
}

// ---------------- epilogue ----------------
__global__ void final_kernel(const Scratch* __restrict__ sc, float* __restrict__ out) {
    double sumterms = 0.0; int nvalid = 0;
    for (int c = 0; c < NC; ++c) {
        bool valid = (sc->cnt_s[c] >= 2u) && (sc->cnt_t[c] >= 2u);
        double t1 = sc->trss[c] + EPSD;
        double t2 = sc->trtt[c] + EPSD;
        double tm = sc->trss[c] + sc->trtt[c] + EPSD;
        double h2ss = -log2(sc->S1[c] / (t1 * t1) + EPSD);
        double h2tt = -log2(sc->S2[c] / (t2 * t2) + EPSD);
        double h2m  = -log2((sc->S1[c] + sc->S2[c] + sc->S3[c]) / (tm * tm) + EPSD);
        double term = h2m - 0.5 * (h2ss + h2tt);
        if (valid) { sumterms += term; ++nvalid; }
    }
    double creda = (nvalid > 0) ? sumterms / (double)nvalid : 0.0;
    double lcls = sc->cls_sum / (double)NS;
    double lent = sc->ent_sum / (double)NT;
    out[0] = (float)(lcls + 1.0 * creda + 0.1 * lent);
}

extern "C" void kernel_launch(void* const* d_in, const int* in_sizes, int n_in,
                              void* d_out, int out_size, void* d_ws, size_t ws_size,
                              hipStream_t stream) {
    const float* fs  = (const float*)d_in[0];  // features_s [1024,2048]
    const float* ls  = (const float*)d_in[1];  // logits_s   [1024,4]
    const int*   lab = (const int*)  d_in[2];  // labels_s   [1024]
    const float* ft  = (const float*)d_in[3];  // features_t [1024,2048]
    const float* lt  = (const float*)d_in[4];  // logits_t   [1024,4]

    float* D2   = (float*)d_ws;                                  // 16 MB
    Scratch* sc = (Scratch*)((char*)d_ws + (size_t)NN * NN * sizeof(float));

    init_kernel<<<1, 32, 0, stream>>>(sc);
    row_kernel<<<(NN + 255) / 256, 256, 0, stream>>>(ls, lab, lt, sc);
    rownorm_kernel<<<NN, 256, 0, stream>>>(fs, ft, sc);
    gram_kernel<<<132, 128, 0, stream>>>(fs, ft, D2, sc);        // 528 wave-tiles (bi<=bj)
    med_setup_kernel<<<1, NC, 0, stream>>>(sc);
    for (int it = 0; it < 32; ++it) {
        med_count_kernel<<<(NN * NN) / 256, 256, 0, stream>>>(D2, sc);
        med_update_kernel<<<1, NC, 0, stream>>>(sc);
    }
    med_final_kernel<<<1, NC, 0, stream>>>(sc);
    sum_kernel<<<(NN * NN) / 256, 256, 0, stream>>>(D2, sc);
    final_kernel<<<1, 1, 0, stream>>>(sc, (float*)d_out);
}